// PhiMHADecoderLayer_27882927685972
// MI455X (gfx1250) — compile-verified
//
#include <hip/hip_runtime.h>
#include <hip/hip_bf16.h>

// ---------------------------------------------------------------------------
// Phi decoder layer for gfx1250 (MI455X), fp32 WMMA (V_WMMA_F32_16X16X4_F32)
// ---------------------------------------------------------------------------

typedef float v2f __attribute__((ext_vector_type(2)));
typedef float v8f __attribute__((ext_vector_type(8)));

#define B_   2
#define S_   1024
#define D_   2048
#define H_   32
#define HK_  8
#define HD_  64
#define RD_  32
#define FF_  8192
#define QKVN ((H_ + 2 * HK_) * HD_)   // 3072

static __device__ __forceinline__ v8f wmma_f32(v2f a, v2f b, v8f c) {
  return __builtin_amdgcn_wmma_f32_16x16x4_f32(false, a, false, b, (short)0, c,
                                               false, false);
}

// ---------------------------------------------------------------------------
// LayerNorm: h = (x - mu) * rsqrt(var + eps) * w + b   (one block per row)
// ---------------------------------------------------------------------------
__global__ __launch_bounds__(256) void ln_kernel(const float* __restrict__ x,
                                                 const float* __restrict__ w,
                                                 const float* __restrict__ bb,
                                                 float* __restrict__ h) {
  const int row = blockIdx.x;           // 0 .. B*S-1
  const int tid = threadIdx.x;
  const float* xr = x + (size_t)row * D_;
  float* hr = h + (size_t)row * D_;

  __shared__ float red[256];
  float v[8];
  float s = 0.f;
#pragma unroll
  for (int i = 0; i < 8; ++i) {
    v[i] = xr[tid + 256 * i];
    s += v[i];
  }
  red[tid] = s;
  __syncthreads();
  for (int off = 128; off > 0; off >>= 1) {
    if (tid < off) red[tid] += red[tid + off];
    __syncthreads();
  }
  const float mean = red[0] * (1.f / (float)D_);
  __syncthreads();

  s = 0.f;
#pragma unroll
  for (int i = 0; i < 8; ++i) {
    float d = v[i] - mean;
    s += d * d;
  }
  red[tid] = s;
  __syncthreads();
  for (int off = 128; off > 0; off >>= 1) {
    if (tid < off) red[tid] += red[tid + off];
    __syncthreads();
  }
  const float rstd = rsqrtf(red[0] * (1.f / (float)D_) + 1e-5f);

#pragma unroll
  for (int i = 0; i < 8; ++i) {
    int c = tid + 256 * i;
    hr[c] = (v[i] - mean) * rstd * w[c] + bb[c];
  }
}

// ---------------------------------------------------------------------------
// Tiled WMMA GEMM: C[M,N] = A[M,K] @ W[N,K]^T + bias, optional epilogue.
// Block = 128 threads = 4 waves. Block tile 64x64, wave tile 16x64.
// K staged in LDS chunks of 32, padded stride 36 (16B-aligned, conflict-free).
// Next K-chunk prefetched into cache (global_prefetch_b8) while WMMAs run.
// EPI: 0 = bias, 1 = bias+gelu(tanh), 2 = bias + res1 + res2
// ---------------------------------------------------------------------------
template <int EPI>
__global__ __launch_bounds__(128) void gemm_kernel(
    const float* __restrict__ A, const float* __restrict__ W,
    const float* __restrict__ bias, float* __restrict__ C, int M, int N, int K,
    const float* __restrict__ res1, const float* __restrict__ res2) {
  __shared__ float As[64][36];
  __shared__ float Ws[64][36];

  const int tid = threadIdx.x;
  const int lane = tid & 31;
  const int wave = tid >> 5;       // 0..3
  const int hf = lane >> 4;        // half-wave: 0 or 1
  const int l16 = lane & 15;
  const int koff = hf * 2;

  const int n0 = blockIdx.x * 64;
  const int m0 = blockIdx.y * 64;

  v8f acc[4];
#pragma unroll
  for (int j = 0; j < 4; ++j)
    acc[j] = (v8f){0.f, 0.f, 0.f, 0.f, 0.f, 0.f, 0.f, 0.f};

  for (int k0 = 0; k0 < K; k0 += 32) {
    __syncthreads();
// cooperative load: 64x32 fp32 for A and W (4 x float4 per thread each)
#pragma unroll
    for (int q = 0; q < 4; ++q) {
      int idx = tid + q * 128;       // 0..511 float4 slots
      int row = idx >> 3;            // 0..63
      int kq = idx & 7;              // 0..7 -> 4 floats each
      float4 av = *(const float4*)(A + (size_t)(m0 + row) * K + k0 + kq * 4);
      *(float4*)(&As[row][kq * 4]) = av;
      float4 wv = *(const float4*)(W + (size_t)(n0 + row) * K + k0 + kq * 4);
      *(float4*)(&Ws[row][kq * 4]) = wv;
    }
    // prefetch next K chunk (one cacheline per tile row; branchless pointer
    // select keeps EXEC uniform around the WMMA block)
    if (k0 + 32 < K) {
      const int r = tid & 63;
      const float* p = (tid < 64) ? (A + (size_t)(m0 + r) * K + k0 + 32)
                                  : (W + (size_t)(n0 + r) * K + k0 + 32);
      __builtin_prefetch(p, 0, 0);
    }
    __syncthreads();

#pragma unroll
    for (int kk = 0; kk < 8; ++kk) {
      const int kb = kk * 4 + koff;
      v2f a;
      a.x = As[wave * 16 + l16][kb];
      a.y = As[wave * 16 + l16][kb + 1];
#pragma unroll
      for (int j = 0; j < 4; ++j) {
        v2f b;
        b.x = Ws[j * 16 + l16][kb];
        b.y = Ws[j * 16 + l16][kb + 1];
        acc[j] = wmma_f32(a, b, acc[j]);   // 4 independent chains interleave
      }
    }
  }

// epilogue
#pragma unroll
  for (int j = 0; j < 4; ++j) {
    const int n = n0 + j * 16 + l16;
#pragma unroll
    for (int r = 0; r < 8; ++r) {
      const int m = m0 + wave * 16 + r + hf * 8;
      float v = acc[j][r] + bias[n];
      if constexpr (EPI == 1) {
        float t = 0.7978845608028654f * (v + 0.044715f * v * v * v);
        v = 0.5f * v * (1.f + tanhf(t));
      }
      if constexpr (EPI == 2) {
        v += res1[(size_t)m * N + n] + res2[(size_t)m * N + n];
      }
      C[(size_t)m * N + n] = v;
    }
  }
}

// ---------------------------------------------------------------------------
// RoPE applied in place to q (H heads) and k (HK heads) inside qkv buffer.
// one thread per (b, s, head, freq-index j in 0..RD/2)
// ---------------------------------------------------------------------------
__global__ __launch_bounds__(256) void rope_kernel(float* __restrict__ qkv) {
  const int total = B_ * S_ * (H_ + HK_) * (RD_ / 2);
  int gid = blockIdx.x * 256 + threadIdx.x;
  if (gid >= total) return;
  const int j = gid & 15;
  int tmp = gid >> 4;
  const int head = tmp % (H_ + HK_);
  tmp /= (H_ + HK_);
  const int s = tmp % S_;
  const int b = tmp / S_;

  const int col = (head < H_) ? head * HD_ : H_ * HD_ + (head - H_) * HD_;
  float* base = qkv + ((size_t)(b * S_ + s)) * QKVN + col;

  const float inv = __powf(10000.f, -(float)j / 16.f);
  const float ang = (float)s * inv;
  const float c = __cosf(ang), sn = __sinf(ang);
  const float u1 = base[j];
  const float u2 = base[j + 16];
  base[j] = u1 * c - u2 * sn;
  base[j + 16] = u1 * sn + u2 * c;
}

// ---------------------------------------------------------------------------
// Causal GQA flash attention. One wave per (b, head, 16-row q tile).
// scores:  S = Q(16xHD) * K^T  via 16 WMMAs, 2 independent accumulator chains
// ctx   : ctx += P(16x16) * V  via 16 WMMAs, 4 chains, A-frag reused over dt
// ---------------------------------------------------------------------------
__global__ __launch_bounds__(128) void attn_kernel(const float* __restrict__ qkv,
                                                   float* __restrict__ ctx) {
  // per-wave LDS: K[16][68] + V[16][68] + P[16][20]  (floats)
  __shared__ float smem[4 * (16 * 68 * 2 + 16 * 20)];

  const int tid = threadIdx.x;
  const int lane = tid & 31;
  const int wv = tid >> 5;
  const int hf = lane >> 4;
  const int l16 = lane & 15;
  const int koff = hf * 2;

  const int wid = blockIdx.x * 4 + wv;        // 0..4095
  const int qt = wid & 63;                    // q tile (S/16 = 64)
  const int hq = (wid >> 6) & 31;             // head
  const int b = wid >> 11;                    // batch
  const int kh = hq >> 2;                     // GQA: rep = H/HK = 4

  float* Ks = smem + wv * (16 * 68 * 2 + 16 * 20);
  float* Vs = Ks + 16 * 68;
  float* Ps = Vs + 16 * 68;

  const int qcol = hq * HD_;
  const int kcol = H_ * HD_ + kh * HD_;
  const int vcol = H_ * HD_ + HK_ * HD_ + kh * HD_;
  const float scale = 0.125f;                 // 1/sqrt(64)

  // preload Q fragments into registers: qreg[2*kc+t] = Q[l16][kc*4+koff+t]
  float qreg[32];
  {
    const float* qrow =
        qkv + ((size_t)(b * S_ + qt * 16 + l16)) * QKVN + qcol;
#pragma unroll
    for (int kc = 0; kc < 16; ++kc) {
      float2 q2 = *(const float2*)(qrow + kc * 4 + koff);
      qreg[2 * kc] = q2.x;
      qreg[2 * kc + 1] = q2.y;
    }
  }

  v8f cacc[4];
#pragma unroll
  for (int dt = 0; dt < 4; ++dt)
    cacc[dt] = (v8f){0.f, 0.f, 0.f, 0.f, 0.f, 0.f, 0.f, 0.f};
  float Mrow[8], Lrow[8];
#pragma unroll
  for (int r = 0; r < 8; ++r) {
    Mrow[r] = -1e30f;
    Lrow[r] = 0.f;
  }

  for (int kt = 0; kt <= qt; ++kt) {
    // ---- stage K and V tiles (16 x 64) into per-wave LDS ----
    const size_t rowbase = (size_t)(b * S_ + kt * 16);
#pragma unroll
    for (int t = 0; t < 8; ++t) {
      int idx = lane + t * 32;       // 0..255 float4 slots
      int row = idx >> 4;
      int c4 = idx & 15;
      float4 kv4 =
          *(const float4*)(qkv + (rowbase + row) * QKVN + kcol + c4 * 4);
      *(float4*)(Ks + row * 68 + c4 * 4) = kv4;
      float4 vv4 =
          *(const float4*)(qkv + (rowbase + row) * QKVN + vcol + c4 * 4);
      *(float4*)(Vs + row * 68 + c4 * 4) = vv4;
    }
    asm volatile("s_wait_dscnt 0" ::: "memory");
    __builtin_amdgcn_wave_barrier();

    // ---- scores tile (16 q x 16 k): two independent WMMA chains ----
    v8f s0 = (v8f){0.f, 0.f, 0.f, 0.f, 0.f, 0.f, 0.f, 0.f};
    v8f s1 = (v8f){0.f, 0.f, 0.f, 0.f, 0.f, 0.f, 0.f, 0.f};
#pragma unroll
    for (int kc = 0; kc < 16; kc += 2) {
      v2f a0, a1;
      a0.x = qreg[2 * kc];
      a0.y = qreg[2 * kc + 1];
      a1.x = qreg[2 * kc + 2];
      a1.y = qreg[2 * kc + 3];
      float2 k0 = *(const float2*)(Ks + l16 * 68 + kc * 4 + koff);
      float2 k1 = *(const float2*)(Ks + l16 * 68 + (kc + 1) * 4 + koff);
      v2f b0, b1;
      b0.x = k0.x; b0.y = k0.y;
      b1.x = k1.x; b1.y = k1.y;
      s0 = wmma_f32(a0, b0, s0);
      s1 = wmma_f32(a1, b1, s1);
    }
    v8f s;
#pragma unroll
    for (int r = 0; r < 8; ++r) s[r] = s0[r] + s1[r];

    // ---- online softmax (row m = r + 8*hf, col n = l16) ----
    float alpha_arr[8];
#pragma unroll
    for (int r = 0; r < 8; ++r) {
      const int m = r + 8 * hf;
      float val = s[r] * scale;
      if (kt == qt && l16 > m) val = -1e30f;   // causal mask (diagonal tile)
      float rm = val;
      rm = fmaxf(rm, __shfl_xor(rm, 1, 32));
      rm = fmaxf(rm, __shfl_xor(rm, 2, 32));
      rm = fmaxf(rm, __shfl_xor(rm, 4, 32));
      rm = fmaxf(rm, __shfl_xor(rm, 8, 32));
      const float mnew = fmaxf(Mrow[r], rm);
      const float p = __expf(val - mnew);
      float ps = p;
      ps += __shfl_xor(ps, 1, 32);
      ps += __shfl_xor(ps, 2, 32);
      ps += __shfl_xor(ps, 4, 32);
      ps += __shfl_xor(ps, 8, 32);
      const float alpha = __expf(Mrow[r] - mnew);
      Lrow[r] = Lrow[r] * alpha + ps;
      Mrow[r] = mnew;
      alpha_arr[r] = alpha;
      Ps[m * 20 + l16] = p;                    // P -> LDS for A-frag relayout
    }
#pragma unroll
    for (int dt = 0; dt < 4; ++dt)
#pragma unroll
      for (int r = 0; r < 8; ++r) cacc[dt][r] *= alpha_arr[r];

    asm volatile("s_wait_dscnt 0" ::: "memory");
    __builtin_amdgcn_wave_barrier();

    // ---- ctx += P * V : kc outer so the 4 dt-chains interleave and the
    //      P A-fragment is reused across 4 WMMAs ----
#pragma unroll
    for (int kc = 0; kc < 4; ++kc) {
      float2 p2 = *(const float2*)(Ps + l16 * 20 + kc * 4 + koff);
      v2f a;
      a.x = p2.x;
      a.y = p2.y;
#pragma unroll
      for (int dt = 0; dt < 4; ++dt) {
        v2f bb;
        bb.x = Vs[(kc * 4 + koff) * 68 + dt * 16 + l16];
        bb.y = Vs[(kc * 4 + koff + 1) * 68 + dt * 16 + l16];
        cacc[dt] = wmma_f32(a, bb, cacc[dt]);
      }
    }
    asm volatile("s_wait_dscnt 0" ::: "memory");
    __builtin_amdgcn_wave_barrier();
  }

  // ---- normalize and write ctx tile ----
#pragma unroll
  for (int dt = 0; dt < 4; ++dt) {
#pragma unroll
    for (int r = 0; r < 8; ++r) {
      const int m = r + 8 * hf;
      const float out = cacc[dt][r] / Lrow[r];
      ctx[((size_t)(b * S_ + qt * 16 + m)) * (H_ * HD_) + hq * HD_ + dt * 16 +
          l16] = out;
    }
  }
}

// ---------------------------------------------------------------------------
// Launcher
// ---------------------------------------------------------------------------
extern "C" void kernel_launch(void* const* d_in, const int* in_sizes, int n_in,
                              void* d_out, int out_size, void* d_ws,
                              size_t ws_size, hipStream_t stream) {
  (void)in_sizes; (void)n_in; (void)out_size; (void)ws_size;
  const float* x     = (const float*)d_in[0];
  const float* ln_w  = (const float*)d_in[1];
  const float* ln_b  = (const float*)d_in[2];
  const float* wqkv  = (const float*)d_in[3];
  const float* bqkv  = (const float*)d_in[4];
  const float* wout  = (const float*)d_in[5];
  const float* bout  = (const float*)d_in[6];
  const float* fc1_w = (const float*)d_in[7];
  const float* fc1_b = (const float*)d_in[8];
  const float* fc2_w = (const float*)d_in[9];
  const float* fc2_b = (const float*)d_in[10];
  float* out = (float*)d_out;

  const size_t M = (size_t)B_ * S_;           // 2048 rows
  float* ws   = (float*)d_ws;
  float* h    = ws;                           // M x D
  float* qkv  = h + M * D_;                   // M x 3072
  float* ctx  = qkv + M * QKVN;               // M x 2048
  float* attn = ctx + M * (H_ * HD_);         // M x 2048
  float* mlp1 = attn + M * D_;                // M x 8192

  // 1. LayerNorm
  ln_kernel<<<(int)M, 256, 0, stream>>>(x, ln_w, ln_b, h);

  // 2. QKV projection: qkv = h @ wqkv^T + bqkv
  gemm_kernel<0><<<dim3(QKVN / 64, (int)M / 64), 128, 0, stream>>>(
      h, wqkv, bqkv, qkv, (int)M, QKVN, D_, nullptr, nullptr);

  // 3. RoPE on q and k heads (in place)
  {
    const int total = B_ * S_ * (H_ + HK_) * (RD_ / 2);
    rope_kernel<<<(total + 255) / 256, 256, 0, stream>>>(qkv);
  }

  // 4. Causal GQA attention -> ctx
  attn_kernel<<<(B_ * H_ * (S_ / 16)) / 4, 128, 0, stream>>>(qkv, ctx);

  // 5. Output projection: attn = ctx @ wout^T + bout
  gemm_kernel<0><<<dim3(D_ / 64, (int)M / 64), 128, 0, stream>>>(
      ctx, wout, bout, attn, (int)M, D_, H_ * HD_, nullptr, nullptr);

  // 6. MLP fc1 + gelu: mlp1 = gelu(h @ fc1_w^T + fc1_b)
  gemm_kernel<1><<<dim3(FF_ / 64, (int)M / 64), 128, 0, stream>>>(
      h, fc1_w, fc1_b, mlp1, (int)M, FF_, D_, nullptr, nullptr);

  // 7. MLP fc2 + residuals: out = mlp1 @ fc2_w^T + fc2_b + attn + x
  gemm_kernel<2><<<dim3(D_ / 64, (int)M / 64), 128, 0, stream>>>(
      mlp1, fc2_w, fc2_b, out, (int)M, D_, FF_, attn, x);
}